// GeneralizedNormPooling_46815143526536
// MI455X (gfx1250) — compile-verified
//
#include <hip/hip_runtime.h>

#define NFEAT 128
#define NSEG  1024
#define EPSF  1e-6f

#define NODES_PER_WAVE  128
#define WAVES_PER_BLOCK 4
#define NODES_PER_BLOCK (NODES_PER_WAVE * WAVES_PER_BLOCK)

typedef __attribute__((ext_vector_type(2))) float v2f;
typedef __attribute__((ext_vector_type(8))) float v8f;

// ---------------------------------------------------------------------------
// Kernel 1: compute device-side scalars (ps0, ps1, 1/ps0, 1/ps1, qs0) and
// zero the segment accumulators + counts in workspace.
// ---------------------------------------------------------------------------
__global__ void gnp_init_kernel(const float* __restrict__ ps_raw,
                                const float* __restrict__ qs_raw,
                                float* __restrict__ scal,
                                float* __restrict__ acc,
                                float* __restrict__ cnts) {
    int i = blockIdx.x * blockDim.x + threadIdx.x;
    if (i < NSEG * NFEAT) acc[i] = 0.0f;
    if (i < NSEG)         cnts[i] = 0.0f;
    if (i == 0) {
        // ps = -Threshold(-50,50)(-softplus(ps_raw)) == softplus clamped
        float ps[2];
        #pragma unroll
        for (int j = 0; j < 2; ++j) {
            float v  = ps_raw[j];
            float sp = (v > 30.0f) ? v : log1pf(expf(v));   // softplus
            ps[j] = (sp < 50.0f) ? sp : -50.0f;
        }
        float qs0 = tanhf(qs_raw[0]);
        scal[0] = ps[0];
        scal[1] = ps[1];
        scal[2] = 1.0f / ps[0];
        scal[3] = 1.0f / ps[1];
        scal[4] = qs0;
    }
}

// ---------------------------------------------------------------------------
// Kernel 2: streaming pass over x (256 MB).  Block = 128 threads = 4 wave32s.
// Each wave owns 128 contiguous nodes; lane L owns features [4L, 4L+3]
// (float4 -> one global_load_b128 per lane per node, wave covers the full
// 512-byte feature row).  `batch` is sorted, so segment ids form runs:
// accumulate (|x|+eps)^p in registers, flush with global f32 atomics only on
// run boundaries (~1-2 flushes per wave).  Lane 0 carries the node count.
// ---------------------------------------------------------------------------
__global__ void gnp_accum_kernel(const float* __restrict__ x,
                                 const long long* __restrict__ batch,
                                 const float* __restrict__ scal,
                                 float* __restrict__ acc,
                                 float* __restrict__ cnts,
                                 int n) {
    __shared__ int sbatch[NODES_PER_BLOCK];
    const int t          = threadIdx.x;
    const int blockStart = blockIdx.x * NODES_PER_BLOCK;

    // Stage this block's segment ids into LDS (int64 -> int, ids < 1024).
    for (int i = t; i < NODES_PER_BLOCK; i += blockDim.x) {
        int node  = blockStart + i;
        sbatch[i] = (node < n) ? (int)batch[node] : -1;
    }
    __syncthreads();

    const int wave = t >> 5;
    const int lane = t & 31;
    const float p  = (lane < 16) ? scal[0] : scal[1];   // feat<64 -> ps0

    const int wBeg = wave * NODES_PER_WAVE;
    const int wEnd = wBeg + NODES_PER_WAVE;

    float sx = 0.0f, sy = 0.0f, sz = 0.0f, sw = 0.0f;
    float cnt = 0.0f;
    int   cur = -1;

    for (int i = wBeg; i < wEnd; ++i) {
        const int node = blockStart + i;
        if (node >= n) break;                 // uniform across the wave
        const int seg = sbatch[i];
        if (seg != cur) {                     // uniform across the wave
            if (cur >= 0) {
                float* a = acc + (size_t)cur * NFEAT + lane * 4;
                atomicAdd(a + 0, sx); atomicAdd(a + 1, sy);
                atomicAdd(a + 2, sz); atomicAdd(a + 3, sw);
                if (lane == 0) atomicAdd(&cnts[cur], cnt);
            }
            sx = sy = sz = sw = 0.0f; cnt = 0.0f; cur = seg;
        }
        const float4 xv =
            *(const float4*)(x + (size_t)node * NFEAT + lane * 4);
        // prefetch a future row (gfx1250 global_prefetch_b8)
        if (node + 32 < n)
            __builtin_prefetch(x + (size_t)(node + 32) * NFEAT + lane * 4, 0, 0);
        sx += __expf(p * __logf(fabsf(xv.x) + EPSF));
        sy += __expf(p * __logf(fabsf(xv.y) + EPSF));
        sz += __expf(p * __logf(fabsf(xv.z) + EPSF));
        sw += __expf(p * __logf(fabsf(xv.w) + EPSF));
        if (lane == 0) cnt += 1.0f;
    }
    if (cur >= 0) {
        float* a = acc + (size_t)cur * NFEAT + lane * 4;
        atomicAdd(a + 0, sx); atomicAdd(a + 1, sy);
        atomicAdd(a + 2, sz); atomicAdd(a + 3, sw);
        if (lane == 0) atomicAdd(&cnts[cur], cnt);
    }
}

// ---------------------------------------------------------------------------
// Kernel 3: gnp[g,f] = s^(1/p) / n^qs0   (1024x128 elements)
// ---------------------------------------------------------------------------
__global__ void gnp_finalize_kernel(const float* __restrict__ scal,
                                    const float* __restrict__ acc,
                                    const float* __restrict__ cnts,
                                    float* __restrict__ gnp) {
    int i = blockIdx.x * blockDim.x + threadIdx.x;
    if (i >= NSEG * NFEAT) return;
    const int f = i & (NFEAT - 1);
    const int g = i >> 7;
    const float invp = (f < 64) ? scal[2] : scal[3];
    const float qs0  = scal[4];
    const float s    = acc[i];
    const float nsz  = cnts[g];
    float v = __expf(invp * __logf(s));          // s^(1/p)
    v *= __expf(-qs0 * __logf(nsz));             // / n^qs0
    gnp[i] = v;
}

// ---------------------------------------------------------------------------
// Kernel 4: out = gnp @ W^T + b via V_WMMA_F32_16X16X4_F32 (full fp32).
// 1024x128 = 64x8 tiles of 16x16; one tile per wave; 32 K-steps of 4.
// Grid exactly covers 512 waves so EXEC is all-1s (WMMA requirement).
//
// A fragment (16x4 f32, ISA 7.12.2): lanes 0-15 hold M=lane, K={0,1};
//   lanes 16-31 hold M=lane-16, K={2,3}  (2 VGPRs per lane).
// B fragment (4x16): mirrored — lanes hold N=lane%16, K split by lane half.
// D (8 VGPRs): VGPR r = row (r + 8*(lane>=16)), col = lane%16.
// ---------------------------------------------------------------------------
__global__ void gnp_gemm_wmma_kernel(const float* __restrict__ A,   // gnp 1024x128
                                     const float* __restrict__ W,   // 128x128
                                     const float* __restrict__ bias,
                                     float* __restrict__ out) {
    const int wid   = (blockIdx.x * blockDim.x + threadIdx.x) >> 5; // 0..511
    const int lane  = threadIdx.x & 31;
    const int tileM = wid >> 3;          // 0..63
    const int tileN = wid & 7;           // 0..7

    const int half = lane >> 4;          // 0 or 1 -> K offset 0 or 2
    const int l16  = lane & 15;
    const int rowA = tileM * 16 + l16;   // M index this lane loads for A
    const int rowW = tileN * 16 + l16;   // N index this lane loads for B=W^T

    v8f c = {};
    #pragma unroll
    for (int kk = 0; kk < 32; ++kk) {
        const int k = kk * 4 + half * 2;
        v2f a, bm;
        a.x  = A[(size_t)rowA * NFEAT + k];
        a.y  = A[(size_t)rowA * NFEAT + k + 1];
        bm.x = W[(size_t)rowW * NFEAT + k];      // B[k][n] = W[n][k]
        bm.y = W[(size_t)rowW * NFEAT + k + 1];
        c = __builtin_amdgcn_wmma_f32_16x16x4_f32(
                /*neg_a=*/false, a, /*neg_b=*/false, bm,
                /*c_mod=*/(short)0, c, /*reuse_a=*/false, /*reuse_b=*/false);
    }

    const float bb = bias[tileN * 16 + l16];
    #pragma unroll
    for (int r = 0; r < 8; ++r) {
        const int row = tileM * 16 + r + half * 8;
        out[(size_t)row * NFEAT + tileN * 16 + l16] = c[r] + bb;
    }
}

// ---------------------------------------------------------------------------
extern "C" void kernel_launch(void* const* d_in, const int* in_sizes, int n_in,
                              void* d_out, int out_size, void* d_ws, size_t ws_size,
                              hipStream_t stream) {
    const float*     x      = (const float*)d_in[0];
    const long long* batch  = (const long long*)d_in[1];
    const float*     ps_raw = (const float*)d_in[2];
    const float*     qs_raw = (const float*)d_in[3];
    const float*     W      = (const float*)d_in[4];
    const float*     bias   = (const float*)d_in[5];
    float*           out    = (float*)d_out;
    const int n = in_sizes[0] / NFEAT;   // 500000 nodes

    // workspace layout (floats): [16 scalars][B*D acc][B counts][B*D gnp]
    float* ws   = (float*)d_ws;
    float* scal = ws;
    float* acc  = ws + 16;
    float* cnts = acc + NSEG * NFEAT;
    float* gnp  = cnts + NSEG;

    gnp_init_kernel<<<(NSEG * NFEAT + 255) / 256, 256, 0, stream>>>(
        ps_raw, qs_raw, scal, acc, cnts);

    const int nblocks = (n + NODES_PER_BLOCK - 1) / NODES_PER_BLOCK;
    gnp_accum_kernel<<<nblocks, WAVES_PER_BLOCK * 32, 0, stream>>>(
        x, batch, scal, acc, cnts, n);

    gnp_finalize_kernel<<<(NSEG * NFEAT + 255) / 256, 256, 0, stream>>>(
        scal, acc, cnts, gnp);

    // 512 waves total: 64 blocks x 256 threads (8 wave32s each)
    gnp_gemm_wmma_kernel<<<64, 256, 0, stream>>>(gnp, W, bias, out);
}